// BipartiteGCN_4741643895565
// MI455X (gfx1250) — compile-verified
//
#include <hip/hip_runtime.h>
#include <hip/hip_bf16.h>

typedef __attribute__((ext_vector_type(2))) float v2f;
typedef __attribute__((ext_vector_type(8))) float v8f;

#define DIMD 128
#define ROWS_PER_BLOCK 64
#define KCHUNK 64
#define APAD 68      // padded A-tile row stride (floats): banks 4m+c, conflict-free b64 reads
#define PSTRIDE 288  // interleaved W pair-row stride (floats): 288%64==32 -> lane halves hit disjoint banks

// Y[M x 128] = X[M x 128] @ W[128 x 128] + bias[128]
// One 128-thread block (4 waves) computes 64 rows x 128 cols.
// Each wave: 16 rows x 128 cols = 8 WMMA f32 16x16x4 accumulator tiles.
// W chunk is staged in LDS row-pair interleaved: WsP[p][n][2] = {W[2p][n], W[2p+1][n]},
// so each B fragment is ONE aligned ds_load_b64 (no register repacking).
__global__ __launch_bounds__(128)
void gcn_gemm_bias_kernel(const float* __restrict__ X, const float* __restrict__ W,
                          const float* __restrict__ bias, float* __restrict__ Y, int M)
{
    __shared__ float WsP[(KCHUNK / 2) * PSTRIDE];  // 36 KB: 32 row-pairs, interleaved
    __shared__ float As[ROWS_PER_BLOCK * APAD];    // 17 KB: padded A tile

    const int tid   = threadIdx.x;
    const int lane  = tid & 31;
    const int wave  = tid >> 5;          // 0..3
    const int m0    = wave * 16;         // wave's row offset inside block
    const int mm    = lane & 15;
    const int halfK = (lane >> 4) * 2;   // 0 for lanes 0-15, 2 for lanes 16-31
    const int row0  = blockIdx.x * ROWS_PER_BLOCK;

    v8f acc[8] = {};

    for (int kb = 0; kb < DIMD; kb += KCHUNK) {
        // cooperative load of W chunk, interleaving row pairs:
        // task j -> pair p = j>>5, col-group c4 = j&31 (cols 4*c4 .. 4*c4+3)
        {
#pragma unroll
            for (int i = 0; i < 8; ++i) {
                int j  = tid + i * 128;
                int p  = j >> 5;
                int c4 = j & 31;
                float4 g0 = *(const float4*)(W + (size_t)(kb + 2 * p)     * DIMD + c4 * 4);
                float4 g1 = *(const float4*)(W + (size_t)(kb + 2 * p + 1) * DIMD + c4 * 4);
                float4 lo = make_float4(g0.x, g1.x, g0.y, g1.y);
                float4 hi = make_float4(g0.z, g1.z, g0.w, g1.w);
                float* dstp = WsP + p * PSTRIDE + c4 * 8;
                *(float4*)(dstp)     = lo;
                *(float4*)(dstp + 4) = hi;
            }
        }
        // cooperative load of A tile (64 rows x 64 cols), zero-fill past M
        {
#pragma unroll
            for (int i = 0; i < 8; ++i) {
                int j  = tid + i * 128;   // float4 index: 16 per row
                int r  = j >> 4;
                int c4 = j & 15;
                int gr = row0 + r;
                float4 v = make_float4(0.f, 0.f, 0.f, 0.f);
                if (gr < M)
                    v = *(const float4*)(X + (size_t)gr * DIMD + kb + c4 * 4);
                *(float4*)(As + r * APAD + c4 * 4) = v;
            }
        }
        __syncthreads();

        const float* aRow = As + (m0 + mm) * APAD;
#pragma unroll
        for (int k = 0; k < KCHUNK; k += 4) {
            // A fragment: {A[m][k+halfK], A[m][k+halfK+1]} -> contiguous, one b64 load
            v2f a = *(const v2f*)(aRow + k + halfK);
            // B fragments: pair index (k+halfK)/2, interleaved -> one b64 load each
            const float* bBase = WsP + ((k + halfK) >> 1) * PSTRIDE + mm * 2;
#pragma unroll
            for (int nt = 0; nt < 8; ++nt) {
                v2f b = *(const v2f*)(bBase + nt * 32);
                acc[nt] = __builtin_amdgcn_wmma_f32_16x16x4_f32(
                    false, a, false, b, (short)0, acc[nt], false, false);
            }
        }
        __syncthreads();
    }

    // epilogue: bias (per-column) + store per documented C/D layout
#pragma unroll
    for (int nt = 0; nt < 8; ++nt) {
        const int col = nt * 16 + mm;
        const float bv = bias[col];
#pragma unroll
        for (int i = 0; i < 8; ++i) {
            int r = row0 + m0 + i + (lane >> 4) * 8;
            if (r < M)
                Y[(size_t)r * DIMD + col] = acc[nt][i] + bv;
        }
    }
}

// acc[dst[e]] += Y[src[e]]  (row-wise, 128 floats). 32 lanes per edge, float4 each.
__global__ __launch_bounds__(256)
void gcn_scatter_add_kernel(const float* __restrict__ Y, const int* __restrict__ src,
                            const int* __restrict__ dst, float* __restrict__ acc, int E)
{
    long t = (long)blockIdx.x * blockDim.x + threadIdx.x;
    int e = (int)(t >> 5);
    int q = (int)(t & 31);
    if (e >= E) return;
    int s = src[e];
    int d = dst[e];
    float4 v = *(const float4*)(Y + (size_t)s * DIMD + q * 4);
    float* p = acc + (size_t)d * DIMD + q * 4;
    atomicAdd(p + 0, v.x);
    atomicAdd(p + 1, v.y);
    atomicAdd(p + 2, v.z);
    atomicAdd(p + 3, v.w);
}

__global__ __launch_bounds__(256)
void gcn_relu_kernel(float* __restrict__ x, long n4)
{
    long t = (long)blockIdx.x * blockDim.x + threadIdx.x;
    if (t >= n4) return;
    float4* p = (float4*)x + t;
    float4 v = *p;
    v.x = fmaxf(v.x, 0.f);
    v.y = fmaxf(v.y, 0.f);
    v.z = fmaxf(v.z, 0.f);
    v.w = fmaxf(v.w, 0.f);
    *p = v;
}

// out[s] = dot(xa[sa[s]], xp[sp[s]]) over D=128. One wave per pair.
__global__ __launch_bounds__(256)
void gcn_dot_kernel(const float* __restrict__ xa, const float* __restrict__ xp,
                    const int* __restrict__ sa, const int* __restrict__ sp,
                    float* __restrict__ out, int S)
{
    long t = (long)blockIdx.x * blockDim.x + threadIdx.x;
    int w = (int)(t >> 5);
    int lane = (int)(t & 31);
    if (w >= S) return;
    int ia = sa[w];
    int ip = sp[w];
    float4 a = *(const float4*)(xa + (size_t)ia * DIMD + lane * 4);
    float4 p = *(const float4*)(xp + (size_t)ip * DIMD + lane * 4);
    float s = a.x * p.x + a.y * p.y + a.z * p.z + a.w * p.w;
#pragma unroll
    for (int off = 16; off > 0; off >>= 1)
        s += __shfl_xor(s, off, 32);
    if (lane == 0) out[w] = s;
}

extern "C" void kernel_launch(void* const* d_in, const int* in_sizes, int n_in,
                              void* d_out, int out_size, void* d_ws, size_t ws_size,
                              hipStream_t stream)
{
    const float* x_author    = (const float*)d_in[0];
    const float* x_paper     = (const float*)d_in[1];
    const int*   edge_author = (const int*)d_in[2];
    const int*   edge_paper  = (const int*)d_in[3];
    const int*   sup_author  = (const int*)d_in[4];
    const int*   sup_paper   = (const int*)d_in[5];
    const float* W_self_a = (const float*)d_in[6];
    const float* b_self_a = (const float*)d_in[7];
    const float* W_self_p = (const float*)d_in[8];
    const float* b_self_p = (const float*)d_in[9];
    const float* W_a2p    = (const float*)d_in[10];
    const float* b_a2p    = (const float*)d_in[11];
    const float* W_p2a    = (const float*)d_in[12];
    const float* b_p2a    = (const float*)d_in[13];

    const int D  = DIMD;
    const int NA = in_sizes[0] / D;
    const int NP = in_sizes[1] / D;
    const int E  = in_sizes[2];
    const int S  = in_sizes[4];
    const int L  = in_sizes[6] / (D * D);

    const size_t szA = (size_t)NA * D;
    const size_t szP = (size_t)NP * D;
    const size_t need = (3 * szA + 3 * szP) * sizeof(float);
    if (ws_size < need) return;  // workspace too small to run

    float* ws     = (float*)d_ws;
    float* bufA_a = ws;
    float* bufA_p = bufA_a + szA;
    float* bufB_a = bufA_p + szP;
    float* bufB_p = bufB_a + szA;
    float* yA     = bufB_p + szP;   // author-side message transform
    float* yP     = yA + szA;       // paper-side message transform

    const float* curA = x_author;
    const float* curP = x_paper;

    const dim3 blk(128);
    const int gA = (NA + ROWS_PER_BLOCK - 1) / ROWS_PER_BLOCK;
    const int gP = (NP + ROWS_PER_BLOCK - 1) / ROWS_PER_BLOCK;
    const long edgeThreads = (long)E * 32;
    const int gScat = (int)((edgeThreads + 255) / 256);

    for (int l = 0; l < L; ++l) {
        float* dstA = (l & 1) ? bufA_a : bufB_a;
        float* dstP = (l & 1) ? bufA_p : bufB_p;
        const size_t wOff = (size_t)l * D * D;
        const size_t bOff = (size_t)l * D;

        // self projections write directly into the accumulation buffers
        gcn_gemm_bias_kernel<<<gA, blk, 0, stream>>>(curA, W_self_a + wOff, b_self_a + bOff, dstA, NA);
        gcn_gemm_bias_kernel<<<gP, blk, 0, stream>>>(curP, W_self_p + wOff, b_self_p + bOff, dstP, NP);
        // message transforms (bias included => each edge contributes +b, matching reference)
        gcn_gemm_bias_kernel<<<gA, blk, 0, stream>>>(curA, W_a2p + wOff, b_a2p + bOff, yA, NA);
        gcn_gemm_bias_kernel<<<gP, blk, 0, stream>>>(curP, W_p2a + wOff, b_p2a + bOff, yP, NP);

        // scatter messages: author->paper and paper->author
        gcn_scatter_add_kernel<<<gScat, 256, 0, stream>>>(yA, edge_author, edge_paper, dstP, E);
        gcn_scatter_add_kernel<<<gScat, 256, 0, stream>>>(yP, edge_paper, edge_author, dstA, E);

        if (l < L - 1) {
            long n4a = (long)(szA / 4), n4p = (long)(szP / 4);
            gcn_relu_kernel<<<(int)((n4a + 255) / 256), 256, 0, stream>>>(dstA, n4a);
            gcn_relu_kernel<<<(int)((n4p + 255) / 256), 256, 0, stream>>>(dstP, n4p);
        }
        curA = dstA;
        curP = dstP;
    }

    const long dotThreads = (long)S * 32;
    gcn_dot_kernel<<<(int)((dotThreads + 255) / 256), 256, 0, stream>>>(
        curA, curP, sup_author, sup_paper, (float*)d_out, S);
}